// DSFDNet2_69045894250530
// MI455X (gfx1250) — compile-verified
//
#include <hip/hip_runtime.h>
#include <cstdint>
#include <cstddef>

// ---------------- problem constants ----------------
#define BATCH      16
#define NPRIOR     136500
#define NSORT      262144          // 2^18 >= NPRIOR, bitonic size (per batch)
#define TOPK       5000
#define CPAD       5008            // TOPK padded to multiple of 16
#define NTILE      313             // CPAD / 16
#define ROWH       314             // halfwords per mask row (pad to even -> u32 aligned)
#define KWORDS     157             // 32-bit keep words per batch (5024 bits)
#define LTILE      4096            // elements per LDS bitonic tile (32 KB of u64)
#define CONF_T     0.01f
#define NMS_T      0.3f
#define PAD_COORD  1.0e30f

typedef __attribute__((ext_vector_type(2))) float v2f;
typedef __attribute__((ext_vector_type(8))) float v8f;
typedef __attribute__((ext_vector_type(4))) int   v4i;

#if defined(__HIP_DEVICE_COMPILE__) && defined(__gfx1250__) && \
    __has_builtin(__builtin_amdgcn_global_load_async_to_lds_b128) && \
    __has_builtin(__builtin_amdgcn_s_wait_asynccnt)
#define USE_ASYNC_LDS 1
typedef __attribute__((address_space(1))) v4i gv4i;   // global 16B packet
typedef __attribute__((address_space(3))) v4i lv4i;   // LDS 16B packet
#endif

// ---------------- kernel 0: zero output ----------------
__global__ void k_zero(float* out, int n) {
    int t = blockIdx.x * blockDim.x + threadIdx.x;
    if (t < n) out[t] = 0.0f;
}

// ---------------- kernel 1: masked scores + sortable keys ----------------
// key = (monotone(score) ^ 0xFFFFFFFF) << 32 | idx  -> ascending sort ==
// descending score with ascending-index tiebreak (matches stable argsort).
__global__ void k_init_keys(const float* __restrict__ conf,
                            unsigned long long* __restrict__ keys,
                            float* __restrict__ scores) {
    int t = blockIdx.x * blockDim.x + threadIdx.x;   // over BATCH*NSORT
    int b = t / NSORT;
    int i = t - b * NSORT;
    unsigned long long key;
    if (i < NPRIOR) {
        float c = conf[((size_t)b * NPRIOR + i) * 2 + 1];
        float s = (c > CONF_T) ? c : -1.0f;
        scores[(size_t)b * NPRIOR + i] = s;
        unsigned int u = __float_as_uint(s);
        unsigned int m = (u & 0x80000000u) ? ~u : (u | 0x80000000u);
        key = ((unsigned long long)(m ^ 0xFFFFFFFFu) << 32) | (unsigned int)i;
    } else {
        key = ((unsigned long long)0xFFFFFFFFu << 32) | (unsigned int)i; // sorts last
    }
    keys[(size_t)b * NSORT + i] = key;
}

// ---------------- kernel 2a: global bitonic step (j >= LTILE only) ----------------
__global__ void k_bitonic(unsigned long long* __restrict__ keys, int j, int k) {
    int t = blockIdx.x * blockDim.x + threadIdx.x;   // over BATCH*NSORT
    int b = t / NSORT;
    int i = t - b * NSORT;
    int ixj = i ^ j;
    if (ixj > i) {
        size_t base = (size_t)b * NSORT;
        unsigned long long a = keys[base + i];
        unsigned long long c = keys[base + ixj];
        bool up = ((i & k) == 0);
        if (up ? (a > c) : (a < c)) {
            keys[base + i]   = c;
            keys[base + ixj] = a;
        }
    }
}

// ---------------- kernel 2b: LDS bitonic (all substages with j <= LTILE/2) ----
// kFirst==2, kLast==LTILE  -> full local sort of each 4096 tile (k=2..4096)
// kFirst==kLast==k (>4096) -> local merge: j = 2048..1 for that k
// Tile inbound copy uses CDNA5 GLOBAL_LOAD_ASYNC_TO_LDS_B128 + s_wait_asynccnt.
__global__ void __launch_bounds__(256)
k_bitonic_local(unsigned long long* __restrict__ keys, int kFirst, int kLast) {
    __shared__ unsigned long long lds[LTILE];
    int t = threadIdx.x;                       // 256 threads
    int base = blockIdx.x * LTILE;             // element index into BATCH*NSORT
    unsigned long long* g = keys + (size_t)base;

#ifdef USE_ASYNC_LDS
    #pragma unroll
    for (int p = 0; p < 8; ++p) {
        int q = p * 256 + t;                   // 16-byte packet index (2 keys)
        __builtin_amdgcn_global_load_async_to_lds_b128(
            (gv4i*)(g + 2 * q), (lv4i*)&lds[2 * q], /*offset=*/0, /*cpol=*/0);
    }
    __builtin_amdgcn_s_wait_asynccnt(0);
    __syncthreads();
#else
    #pragma unroll
    for (int p = 0; p < 16; ++p) { int e = p * 256 + t; lds[e] = g[e]; }
    __syncthreads();
#endif

    for (int k = kFirst; k <= kLast; k <<= 1) {
        int j0 = (k >> 1) < (LTILE / 2) ? (k >> 1) : (LTILE / 2);
        for (int j = j0; j > 0; j >>= 1) {
            #pragma unroll
            for (int p = 0; p < 8; ++p) {
                int q = p * 256 + t;                              // pair id 0..2047
                int i = ((q & ~(j - 1)) << 1) | (q & (j - 1));    // low element
                int gi = (base + i) & (NSORT - 1);                // idx within batch seg
                bool up = ((gi & k) == 0);
                unsigned long long a = lds[i];
                unsigned long long c = lds[i + j];
                if (up ? (a > c) : (a < c)) { lds[i] = c; lds[i + j] = a; }
            }
            __syncthreads();
        }
    }

    #pragma unroll
    for (int p = 0; p < 16; ++p) { int e = p * 256 + t; g[e] = lds[e]; }
}

// ---------------- kernel 3: gather top-K + decode boxes ----------------
__global__ void k_gather_decode(const unsigned long long* __restrict__ keys,
                                const float* __restrict__ scores,
                                const float* __restrict__ loc,
                                const float* __restrict__ prior,
                                float* __restrict__ cx1, float* __restrict__ cy1,
                                float* __restrict__ cx2, float* __restrict__ cy2,
                                float* __restrict__ carea, float* __restrict__ cscore) {
    int t = blockIdx.x * blockDim.x + threadIdx.x;   // over BATCH*CPAD
    int b = t / CPAD;
    int r = t - b * CPAD;
    float x1 = PAD_COORD, y1 = PAD_COORD, x2 = PAD_COORD, y2 = PAD_COORD;
    float area = 0.0f, sc = -1.0f;
    if (r < TOPK) {
        unsigned long long key = keys[(size_t)b * NSORT + r];
        unsigned int idx = (unsigned int)key;
        if (idx < NPRIOR) {
            sc = scores[(size_t)b * NPRIOR + idx];
            const float* l = loc   + ((size_t)b * NPRIOR + idx) * 4;
            const float* p = prior + (size_t)idx * 4;
            float cx = p[0] + l[0] * 0.1f * p[2];
            float cy = p[1] + l[1] * 0.1f * p[3];
            float w  = p[2] * __expf(l[2] * 0.2f);
            float h  = p[3] * __expf(l[3] * 0.2f);
            x1 = cx - 0.5f * w;
            y1 = cy - 0.5f * h;
            x2 = x1 + w;
            y2 = y1 + h;
            area = (x2 - x1) * (y2 - y1);
        }
    }
    int o = b * CPAD + r;
    cx1[o] = x1; cy1[o] = y1; cx2[o] = x2; cy2[o] = y2;
    carea[o] = area; cscore[o] = sc;
}

// ---------------- kernel 4: pairwise suppression mask (WMMA tile) ----------------
// Each wave32 computes a 16x16 tile of the (i,j) pair matrix:
//   areaSum tile via V_WMMA_F32_16X16X4_F32 (rank-2 bilinear term of IoU denom),
//   intersection via VALU, bit = (inter > 0.3*(areaSum - inter)) && (j > i).
__global__ void k_build_mask(const float* __restrict__ cx1, const float* __restrict__ cy1,
                             const float* __restrict__ cx2, const float* __restrict__ cy2,
                             const float* __restrict__ carea,
                             unsigned short* __restrict__ mask16) {
    int jt = blockIdx.x * blockDim.y + threadIdx.y;  // column tile (one wave each)
    if (jt >= NTILE) return;                         // wave-uniform exit, EXEC stays all-1s
    int it = blockIdx.y;
    int b  = blockIdx.z;
    int lane = threadIdx.x;                          // 0..31
    int hf   = lane >> 4;                            // lane half
    int lp   = lane & 15;
    int i0 = it * 16, j0 = jt * 16;
    const int cb = b * CPAD;

    float asum[8];
#if defined(__HIP_DEVICE_COMPILE__) && defined(__gfx1250__) && \
    __has_builtin(__builtin_amdgcn_wmma_f32_16x16x4_f32)
    {
        // A (16x4): lanes 0-15 hold rows M=0..15 for K=0,1 ; lanes 16-31 (K=2,3) = 0
        // B (4x16): lanes 0-15 hold cols N=0..15 for K=0,1 ; lanes 16-31 = 0
        v2f av = {0.0f, 0.0f};
        v2f bv = {0.0f, 0.0f};
        if (hf == 0) {
            av.x = carea[cb + i0 + lp]; av.y = 1.0f;   // row: [area_i, 1]
            bv.x = 1.0f; bv.y = carea[cb + j0 + lp];   // col: [1, area_j]
        }
        v8f cv = {};
        cv = __builtin_amdgcn_wmma_f32_16x16x4_f32(
            /*neg_a=*/false, av, /*neg_b=*/false, bv,
            /*c_mod=*/(short)0, cv, /*reuse_a=*/false, /*reuse_b=*/false);
        #pragma unroll
        for (int r = 0; r < 8; ++r) asum[r] = cv[r];   // C layout: M = r + 8*hf, N = lp
    }
#else
    #pragma unroll
    for (int r = 0; r < 8; ++r)
        asum[r] = carea[cb + i0 + r + 8 * hf] + carea[cb + j0 + lp];
#endif

    // this lane's column box
    float bx1 = cx1[cb + j0 + lp], by1 = cy1[cb + j0 + lp];
    float bx2 = cx2[cb + j0 + lp], by2 = cy2[cb + j0 + lp];
    int gj = j0 + lp;

    #pragma unroll
    for (int r = 0; r < 8; ++r) {
        int gi = i0 + r + 8 * hf;                      // row for this lane's half
        float rx1 = cx1[cb + gi], ry1 = cy1[cb + gi];
        float rx2 = cx2[cb + gi], ry2 = cy2[cb + gi];
        float iw = fminf(rx2, bx2) - fmaxf(rx1, bx1);
        float ih = fminf(ry2, by2) - fmaxf(ry1, by1);
        float inter = fmaxf(iw, 0.0f) * fmaxf(ih, 0.0f);
        float uni = asum[r] - inter;
        bool sup = (inter > NMS_T * uni) && (gj > gi);
        unsigned long long bal = __ballot(sup ? 1 : 0);  // wave32: low 32 bits
        if (lane == 0)
            mask16[((size_t)cb + (i0 + r))     * ROWH + jt] = (unsigned short)(bal & 0xFFFFu);
        if (lane == 16)
            mask16[((size_t)cb + (i0 + r + 8)) * ROWH + jt] = (unsigned short)((bal >> 16) & 0xFFFFu);
    }
}

// ---------------- kernel 5: sequential greedy NMS scan (one WG per batch) ----------------
__global__ void k_greedy_scan(const float* __restrict__ cscore,
                              const unsigned short* __restrict__ mask16,
                              unsigned int* __restrict__ keepW) {
    int b = blockIdx.x;
    int t = threadIdx.x;                               // 256 threads
    __shared__ unsigned int keep[KWORDS];
    if (t < KWORDS) {
        unsigned int wbits = 0;
        for (int bit = 0; bit < 32; ++bit) {
            int j = t * 32 + bit;
            if (j < TOPK && cscore[b * CPAD + j] > CONF_T) wbits |= (1u << bit);
        }
        keep[t] = wbits;
    }
    __syncthreads();
    for (int i = 0; i < TOPK; ++i) {
        unsigned int kw = keep[i >> 5];                // uniform LDS read
        if ((kw >> (i & 31)) & 1u) {                   // uniform branch
            const unsigned int* row =
                (const unsigned int*)(mask16 + ((size_t)(b * CPAD) + i) * ROWH);
            __builtin_prefetch((const void*)(mask16 + ((size_t)(b * CPAD) + i + 1) * ROWH), 0, 1);
            for (int w = t; w < KWORDS; w += blockDim.x)
                keep[w] &= ~row[w];                    // mask has only j>i bits
            __syncthreads();
        }
    }
    if (t < KWORDS) keepW[b * KWORDS + t] = keep[t];
}

// ---------------- kernel 6: compact kept rows into output ----------------
__global__ void k_compact(const unsigned int* __restrict__ keepW,
                          const float* __restrict__ cx1, const float* __restrict__ cy1,
                          const float* __restrict__ cx2, const float* __restrict__ cy2,
                          const float* __restrict__ cscore,
                          float* __restrict__ out) {
    int b = blockIdx.x;
    int t = threadIdx.x;                               // 256 threads
    __shared__ unsigned int words[KWORDS];
    __shared__ unsigned int base[KWORDS];
    if (t < KWORDS) words[t] = keepW[b * KWORDS + t];
    __syncthreads();
    if (t == 0) {
        unsigned int s = 0;
        for (int w = 0; w < KWORDS; ++w) { base[w] = s; s += __popc(words[w]); }
    }
    __syncthreads();
    for (int w = t; w < KWORDS; w += blockDim.x) {
        unsigned int m = words[w];
        while (m) {
            int bit = __ffs(m) - 1;
            m &= m - 1;
            int j = w * 32 + bit;
            unsigned int rank = base[w] + __popc(words[w] & ((1u << bit) - 1u));
            size_t o = (((size_t)b * 2 + 1) * TOPK + rank) * 5;   // class-1 plane
            int c = b * CPAD + j;
            out[o + 0] = cscore[c];
            out[o + 1] = cx1[c];
            out[o + 2] = cy1[c];
            out[o + 3] = cx2[c];
            out[o + 4] = cy2[c];
        }
    }
}

// ---------------- host launcher ----------------
extern "C" void kernel_launch(void* const* d_in, const int* in_sizes, int n_in,
                              void* d_out, int out_size, void* d_ws, size_t ws_size,
                              hipStream_t stream) {
    const float* loc   = (const float*)d_in[0];   // (16, 136500, 4)
    const float* conf  = (const float*)d_in[1];   // (16*136500, 2)
    const float* prior = (const float*)d_in[2];   // (136500, 4)
    float* out = (float*)d_out;                   // (16, 2, 5000, 5)

    // workspace carve-up (all 256B aligned)
    char* w = (char*)d_ws;
    size_t off = 0;
    auto carve = [&](size_t bytes) { char* p = w + off; off = (off + bytes + 255) & ~(size_t)255; return p; };
    unsigned long long* keys = (unsigned long long*)carve((size_t)BATCH * NSORT * 8);   // 33.5 MB
    float*  scores = (float*)carve((size_t)BATCH * NPRIOR * 4);                          // 8.7 MB
    float*  cx1    = (float*)carve((size_t)BATCH * CPAD * 4);
    float*  cy1    = (float*)carve((size_t)BATCH * CPAD * 4);
    float*  cx2    = (float*)carve((size_t)BATCH * CPAD * 4);
    float*  cy2    = (float*)carve((size_t)BATCH * CPAD * 4);
    float*  carea  = (float*)carve((size_t)BATCH * CPAD * 4);
    float*  cscore = (float*)carve((size_t)BATCH * CPAD * 4);
    unsigned int* keepW = (unsigned int*)carve((size_t)BATCH * KWORDS * 4);
    unsigned short* mask16 = (unsigned short*)carve((size_t)BATCH * CPAD * ROWH * 2);    // 50.3 MB
    (void)ws_size; (void)in_sizes; (void)n_in; (void)out_size;

    const int OUTN = BATCH * 2 * TOPK * 5;
    k_zero<<<(OUTN + 255) / 256, 256, 0, stream>>>(out, OUTN);

    const int NK = BATCH * NSORT;                 // 4,194,304
    const int NLB = NK / LTILE;                   // 1024 local-sort blocks
    k_init_keys<<<NK / 256, 256, 0, stream>>>(conf, keys, scores);

    // hybrid bitonic: local sort (k=2..4096 in LDS), then per k>4096:
    // global passes for j>=4096 + one local merge kernel for j=2048..1.
    k_bitonic_local<<<NLB, 256, 0, stream>>>(keys, 2, LTILE);
    for (int k = LTILE * 2; k <= NSORT; k <<= 1) {
        for (int j = k >> 1; j >= LTILE; j >>= 1)
            k_bitonic<<<NK / 256, 256, 0, stream>>>(keys, j, k);
        k_bitonic_local<<<NLB, 256, 0, stream>>>(keys, k, k);
    }

    k_gather_decode<<<(BATCH * CPAD) / 256, 256, 0, stream>>>(
        keys, scores, loc, prior, cx1, cy1, cx2, cy2, carea, cscore);

    dim3 mgrid((NTILE + 3) / 4, NTILE, BATCH);    // 4 waves / block, one 16x16 tile each
    dim3 mblk(32, 4);
    k_build_mask<<<mgrid, mblk, 0, stream>>>(cx1, cy1, cx2, cy2, carea, mask16);

    k_greedy_scan<<<BATCH, 256, 0, stream>>>(cscore, mask16, keepW);

    k_compact<<<BATCH, 256, 0, stream>>>(keepW, cx1, cy1, cx2, cy2, cscore, out);
}